// UniGCNIIConv_29575144800475
// MI455X (gfx1250) — compile-verified
//
#include <hip/hip_runtime.h>

#define D    128
#define NVTX 100000
#define NEDG 20000

typedef __attribute__((ext_vector_type(2))) float v2f;
typedef __attribute__((ext_vector_type(8))) float v8f;

// ---------------------------------------------------------------------------
// 1) zero workspace
// ---------------------------------------------------------------------------
__global__ void zero_f4(float4* __restrict__ p, long n4) {
  long i = (long)blockIdx.x * blockDim.x + threadIdx.x;
  long stride = (long)gridDim.x * blockDim.x;
  float4 z = {0.f, 0.f, 0.f, 0.f};
  for (; i < n4; i += stride) p[i] = z;
}

// ---------------------------------------------------------------------------
// 2,4) wave-per-nnz gather row + scatter atomic-add (used for both stages)
//      src[gidx[i]] row (128 f32) -> atomicAdd into accum[sidx[i]] row
// ---------------------------------------------------------------------------
__global__ __launch_bounds__(256) void scatter_add(
    const float* __restrict__ src, const int* __restrict__ gidx,
    const int* __restrict__ sidx, float* __restrict__ accum,
    float* __restrict__ cnt, int nnz) {
  int warp = (int)((blockIdx.x * blockDim.x + threadIdx.x) >> 5);
  int lane = threadIdx.x & 31;
  if (warp >= nnz) return;
  int g = gidx[warp];
  int s = sidx[warp];
  const float4* row = (const float4*)(src + (long)g * D);
  float4 v = row[lane];                       // 32 lanes x 16B = full row
  float* dst = accum + (long)s * D + lane * 4;
  unsafeAtomicAdd(dst + 0, v.x);
  unsafeAtomicAdd(dst + 1, v.y);
  unsafeAtomicAdd(dst + 2, v.z);
  unsafeAtomicAdd(dst + 3, v.w);
  if (lane == 0) unsafeAtomicAdd(cnt + s, 1.0f);
}

// ---------------------------------------------------------------------------
// 3) in-place divide rows by max(count,1)
// ---------------------------------------------------------------------------
__global__ void div_cnt(float* __restrict__ sum, const float* __restrict__ cnt,
                        long total_f4) {
  long i = (long)blockIdx.x * blockDim.x + threadIdx.x;
  if (i >= total_f4) return;
  long r = i / (D / 4);
  float c = cnt[r];
  c = c > 1.0f ? c : 1.0f;
  float inv = 1.0f / c;
  float4* p = (float4*)sum;
  float4 v = p[i];
  v.x *= inv; v.y *= inv; v.z *= inv; v.w *= inv;
  p[i] = v;
}

// ---------------------------------------------------------------------------
// 5) fused: Xv = Xv_sum/cnt ; Xi = (1-a)Xv + a*X0 ; out = (1-b)Xi + b*(Xi@W^T)
//    16-row tile per block, 8 waves x 16-col WMMA tiles, K=128 via 32 steps
//    of V_WMMA_F32_16X16X4_F32 (fp32, matches reference precision).
// ---------------------------------------------------------------------------
__global__ __launch_bounds__(256) void final_wmma(
    const float* __restrict__ xv_sum, const float* __restrict__ v_cnt,
    const float* __restrict__ x0, const float* __restrict__ W,
    const float* __restrict__ alpha_p, const float* __restrict__ beta_p,
    float* __restrict__ out) {
  __shared__ float xi[16 * 132];              // padded stride: bank-conflict free
  const float alpha = *alpha_p;
  const float beta  = *beta_p;
  const int tid = threadIdx.x;
  const long rowbase = (long)blockIdx.x * 16;

  // build Xi tile: thread -> 8 consecutive floats of one row
  {
    int r  = tid >> 4;                        // 0..15
    int cb = (tid & 15) * 8;                  // 0..120
    long g = rowbase + r;
    float c = v_cnt[g];
    c = c > 1.0f ? c : 1.0f;
    float inv = (1.0f - alpha) / c;
    const float* srow = xv_sum + g * D + cb;
    const float* xrow = x0 + g * D + cb;
    float* drow = xi + r * 132 + cb;
#pragma unroll
    for (int k = 0; k < 8; ++k) drow[k] = srow[k] * inv + alpha * xrow[k];
  }
  __syncthreads();

  const int lane  = tid & 31;
  const int wave  = tid >> 5;
  const int jbase = wave * 16;                // this wave's 16 output columns
  const int mm    = lane & 15;                // matrix row (A) / column (B,D)
  const int khalf = (lane >> 4) * 2;          // lane-half selects K pair

  v8f acc = {};
  const float* wrow = W + (long)(jbase + mm) * D;  // B[k][j] = W[j][k]
#pragma unroll 8
  for (int kt = 0; kt < 32; ++kt) {
    int kk = kt * 4 + khalf;
    v2f a; a.x = xi[mm * 132 + kk]; a.y = xi[mm * 132 + kk + 1];
    v2f b; b.x = wrow[kk];          b.y = wrow[kk + 1];
    acc = __builtin_amdgcn_wmma_f32_16x16x4_f32(
        /*neg_a=*/false, a, /*neg_b=*/false, b,
        /*c_mod=*/(short)0, acc, /*reuse_a=*/false, /*reuse_b=*/false);
  }

  const float obeta = 1.0f - beta;
#pragma unroll
  for (int r = 0; r < 8; ++r) {
    int row = r + ((lane >> 4) << 3);         // C/D layout: lanes 16-31 -> M+8
    int col = jbase + mm;
    long g = rowbase + row;
    out[g * D + col] = obeta * xi[row * 132 + col] + beta * acc[r];
  }
}

// ---------------------------------------------------------------------------
extern "C" void kernel_launch(void* const* d_in, const int* in_sizes, int n_in,
                              void* d_out, int out_size, void* d_ws, size_t ws_size,
                              hipStream_t stream) {
  const float* X      = (const float*)d_in[0];
  const int*   vertex = (const int*)d_in[1];
  const int*   edges  = (const int*)d_in[2];
  const float* alpha  = (const float*)d_in[3];
  const float* beta   = (const float*)d_in[4];
  const float* X0     = (const float*)d_in[5];
  const float* W      = (const float*)d_in[6];
  float* out = (float*)d_out;

  float* ws     = (float*)d_ws;
  float* xe_sum = ws;                          // E*D
  float* xv_sum = xe_sum + (long)NEDG * D;     // N*D
  float* e_cnt  = xv_sum + (long)NVTX * D;     // E
  float* v_cnt  = e_cnt + NEDG;                // N

  const int nnz = in_sizes[1];

  long total_f = (long)NEDG * D + (long)NVTX * D + NEDG + NVTX;
  long n4 = (total_f + 3) / 4;
  zero_f4<<<4096, 256, 0, stream>>>((float4*)d_ws, n4);

  int blocks_nnz = (int)(((long)nnz * 32 + 255) / 256);
  // stage 1: vertex -> edge mean (sum part)
  scatter_add<<<blocks_nnz, 256, 0, stream>>>(X, vertex, edges, xe_sum, e_cnt, nnz);
  long ef4 = (long)NEDG * (D / 4);
  div_cnt<<<(int)((ef4 + 255) / 256), 256, 0, stream>>>(xe_sum, e_cnt, ef4);
  // stage 2: edge -> vertex mean (sum part)
  scatter_add<<<blocks_nnz, 256, 0, stream>>>(xe_sum, edges, vertex, xv_sum, v_cnt, nnz);
  // fused divide + residual mix + fp32 WMMA projection
  final_wmma<<<NVTX / 16, 256, 0, stream>>>(xv_sum, v_cnt, X0, W, alpha, beta, out);
}